// HeteroGNN_22179211116859
// MI455X (gfx1250) — compile-verified
//
#include <hip/hip_runtime.h>
#include <hip/hip_bf16.h>

// ---------------------------------------------------------------------------
// HeteroGNN on MI455X (gfx1250): fp32 WMMA GEMMs + L2-resident edge scatter.
// ---------------------------------------------------------------------------

typedef float v2f __attribute__((ext_vector_type(2)));
typedef float v8f __attribute__((ext_vector_type(8)));

#define N_REGION  50000
#define N_SUBJECT 8192
#define D_REGION  128
#define D_SUBJECT 64
#define HIDDEN    128
#define OUTD      64

// ---------------------------------------------------------------------------
// Utility kernels
// ---------------------------------------------------------------------------
__global__ void zero_f32(float* __restrict__ p, int n) {
    int i = blockIdx.x * 256 + threadIdx.x;
    if (i < n) p[i] = 0.0f;
}

__global__ void count_edges(const int* __restrict__ dst, float* __restrict__ deg, int E) {
    int e = blockIdx.x * 256 + threadIdx.x;
    if (e < E) atomicAdd(&deg[dst[e]], 1.0f);
}

// dinv[i] = rsqrt(deg[i] + 1)   (GCN: in-degree + self loop)
__global__ void compute_dinv(const float* __restrict__ deg, float* __restrict__ dinv, int n) {
    int i = blockIdx.x * 256 + threadIdx.x;
    if (i < n) dinv[i] = rsqrtf(deg[i] + 1.0f);
}

// ---------------------------------------------------------------------------
// fp32 WMMA GEMM:  C[M,N] = A[M,K] @ B[K,N] (+ bias) (+= existing if acc)
// One wave computes one 16x16 output tile via V_WMMA_F32_16X16X4_F32.
// A-frag (16x4):  lane<16 -> M=lane, holds K = k+0,k+1 ; lane>=16 -> K = k+2,k+3
// B-frag (4x16):  lane<16 -> N=lane, holds K = k+0,k+1 ; lane>=16 -> K = k+2,k+3
// C/D (16x16):    vgpr j, lane l -> M = j + 8*(l>=16), N = l&15
// ---------------------------------------------------------------------------
__global__ void gemm_wmma_f32(const float* __restrict__ A,
                              const float* __restrict__ B,
                              const float* __restrict__ bias,
                              float* __restrict__ C,
                              int M, int K, int N, int tilesN, int totalTiles, int acc) {
    int wave = (int)((blockIdx.x * blockDim.x + threadIdx.x) >> 5);
    if (wave >= totalTiles) return;               // wave-uniform
    int lane  = threadIdx.x & 31;
    int tm    = wave / tilesN;
    int tn    = wave - tm * tilesN;
    int row0  = tm << 4;
    int col0  = tn << 4;
    int l15   = lane & 15;
    int khalf = (lane >> 4) << 1;                 // 0 or 2

    int arow = row0 + l15;
    if (arow >= M) arow = M - 1;                  // clamp loads; stores are guarded
    const float* Arow = A + (size_t)arow * K;
    int bcol = col0 + l15;

    v8f c = {};
    for (int k = 0; k < K; k += 4) {
        v2f a, b;
        // two consecutive K values per lane (8B-aligned pair)
        const float2 ap = *(const float2*)(Arow + k + khalf);
        a.x = ap.x; a.y = ap.y;
        b.x = B[(size_t)(k + khalf)     * N + bcol];
        b.y = B[(size_t)(k + khalf + 1) * N + bcol];
        c = __builtin_amdgcn_wmma_f32_16x16x4_f32(
                /*neg_a=*/false, a, /*neg_b=*/false, b,
                /*c_mod=*/(short)0, c, /*reuse_a=*/false, /*reuse_b=*/false);
    }

    float bv = bias ? bias[bcol] : 0.0f;
    int rbase = row0 + ((lane >> 4) << 3);
#pragma unroll
    for (int j = 0; j < 8; ++j) {
        int r = rbase + j;
        if (r < M) {
            size_t idx = (size_t)r * N + bcol;
            float v = c[j] + bv;
            if (acc) v += C[idx];
            C[idx] = v;
        }
    }
}

// ---------------------------------------------------------------------------
// Edge scatter (feature dim fixed at 128): 32 lanes per edge, float4 per lane.
//   use_norm=1:  acc[dst] += h[src] * dinv[src]*dinv[dst]   (GCN)
//   use_norm=0:  acc[dst] += h[src]                          (SAGE sum)
// ---------------------------------------------------------------------------
__global__ void edge_scatter128(const float* __restrict__ h,
                                const int* __restrict__ src,
                                const int* __restrict__ dst,
                                const float* __restrict__ dinv,
                                float* __restrict__ acc,
                                int E, int use_norm) {
    int e = blockIdx.x * 8 + (threadIdx.x >> 5);
    if (e >= E) return;
    int lane = threadIdx.x & 31;
    int s = src[e], d = dst[e];
    float norm = use_norm ? (dinv[s] * dinv[d]) : 1.0f;
    float4 v = ((const float4*)(h + (size_t)s * 128))[lane];
    float* ap = acc + (size_t)d * 128 + lane * 4;
    atomicAdd(ap + 0, v.x * norm);
    atomicAdd(ap + 1, v.y * norm);
    atomicAdd(ap + 2, v.z * norm);
    atomicAdd(ap + 3, v.w * norm);
}

// out[i][c] = acc[i][c] + h[i][c] * (1/(deg[i]+1)) + b[c]   (out may alias acc)
__global__ void gcn_finalize128(const float* __restrict__ acc,
                                const float* __restrict__ h,
                                const float* __restrict__ dinv,
                                const float* __restrict__ b,
                                float* __restrict__ out, int n) {
    int idx = blockIdx.x * 256 + threadIdx.x;
    if (idx >= n * 128) return;
    int node = idx >> 7;
    int c    = idx & 127;
    float di = dinv[node];
    out[idx] = acc[idx] + h[idx] * (di * di) + b[c];
}

// sum[i][c] /= max(cnt[i], 1)
__global__ void mean_div128(float* __restrict__ sum, const float* __restrict__ cnt, int n) {
    int idx = blockIdx.x * 256 + threadIdx.x;
    if (idx >= n * 128) return;
    int node = idx >> 7;
    sum[idx] = sum[idx] / fmaxf(cnt[node], 1.0f);
}

// ---------------------------------------------------------------------------
// Host-side launcher
// ---------------------------------------------------------------------------
static inline void launch_gemm(const float* A, const float* B, const float* bias,
                               float* C, int M, int K, int N, int acc, hipStream_t s) {
    int tilesN = N >> 4;
    int tilesM = (M + 15) >> 4;
    int total  = tilesM * tilesN;
    int blocks = (total + 7) / 8;                 // 8 waves (256 thr) per block
    gemm_wmma_f32<<<blocks, 256, 0, s>>>(A, B, bias, C, M, K, N, tilesN, total, acc);
}

extern "C" void kernel_launch(void* const* d_in, const int* in_sizes, int n_in,
                              void* d_out, int out_size, void* d_ws, size_t ws_size,
                              hipStream_t stream) {
    const float* x_region  = (const float*)d_in[0];
    const float* x_subject = (const float*)d_in[1];
    const int*   rr_src    = (const int*)d_in[2];
    const int*   rr_dst    = (const int*)d_in[3];
    const int*   hs_src    = (const int*)d_in[4];
    const int*   hs_dst    = (const int*)d_in[5];
    const float* Wg1  = (const float*)d_in[6];
    const float* bg1  = (const float*)d_in[7];
    const float* Wl1  = (const float*)d_in[8];
    const float* bl1  = (const float*)d_in[9];
    const float* Wr1  = (const float*)d_in[10];
    const float* Wg2  = (const float*)d_in[11];
    const float* bg2  = (const float*)d_in[12];
    const float* Wl2  = (const float*)d_in[13];
    const float* bl2  = (const float*)d_in[14];
    const float* Wr2  = (const float*)d_in[15];
    const float* Wout = (const float*)d_in[16];
    const float* bout = (const float*)d_in[17];
    float* out = (float*)d_out;

    const int E_RR = in_sizes[2];
    const int E_HS = in_sizes[4];

    // -------- workspace layout (floats) --------
    float* ws = (float*)d_ws;
    size_t o = 0;
    float* DEG  = ws + o; o += N_REGION;
    float* DINV = ws + o; o += N_REGION;
    float* CNT  = ws + o; o += N_SUBJECT;
    o = (o + 127) & ~(size_t)127;
    float* HBUF = ws + o; o += (size_t)N_REGION * HIDDEN;   // region GEMM result / s2 reuse
    float* ACC  = ws + o; o += (size_t)N_REGION * HIDDEN;   // GCN aggregation / r2 (in-place)
    float* R1   = ws + o; o += (size_t)N_REGION * HIDDEN;   // layer-1 region output
    float* SUM  = ws + o; o += (size_t)N_SUBJECT * HIDDEN;  // SAGE neighbor sum / mean
    float* S1   = ws + o; o += (size_t)N_SUBJECT * HIDDEN;  // layer-1 subject output
    float* S2   = HBUF;                                     // reuse HBUF after r2 is done

    const int NR128 = N_REGION * HIDDEN;   // 6,400,000
    const int NS128 = N_SUBJECT * HIDDEN;  // 1,048,576
    auto blk = [](int n) { return (n + 255) / 256; };

    // -------- degrees / counts --------
    zero_f32<<<blk(N_REGION), 256, 0, stream>>>(DEG, N_REGION);
    zero_f32<<<blk(N_SUBJECT), 256, 0, stream>>>(CNT, N_SUBJECT);
    count_edges<<<blk(E_RR), 256, 0, stream>>>(rr_dst, DEG, E_RR);
    count_edges<<<blk(E_HS), 256, 0, stream>>>(hs_dst, CNT, E_HS);
    compute_dinv<<<blk(N_REGION), 256, 0, stream>>>(DEG, DINV, N_REGION);

    // -------- Layer 1: GCN on region --------
    launch_gemm(x_region, Wg1, nullptr, HBUF, N_REGION, D_REGION, HIDDEN, 0, stream);
    zero_f32<<<blk(NR128), 256, 0, stream>>>(ACC, NR128);
    edge_scatter128<<<(E_RR + 7) / 8, 256, 0, stream>>>(HBUF, rr_src, rr_dst, DINV, ACC, E_RR, 1);
    gcn_finalize128<<<blk(NR128), 256, 0, stream>>>(ACC, HBUF, DINV, bg1, R1, N_REGION);

    // -------- Layer 1: SAGE region -> subject --------
    zero_f32<<<blk(NS128), 256, 0, stream>>>(SUM, NS128);
    edge_scatter128<<<(E_HS + 7) / 8, 256, 0, stream>>>(x_region, hs_src, hs_dst, nullptr, SUM, E_HS, 0);
    mean_div128<<<blk(NS128), 256, 0, stream>>>(SUM, CNT, N_SUBJECT);
    launch_gemm(SUM, Wl1, bl1, S1, N_SUBJECT, HIDDEN, HIDDEN, 0, stream);
    launch_gemm(x_subject, Wr1, nullptr, S1, N_SUBJECT, D_SUBJECT, HIDDEN, 1, stream);

    // -------- Layer 2: GCN on region --------
    launch_gemm(R1, Wg2, nullptr, HBUF, N_REGION, HIDDEN, HIDDEN, 0, stream);
    zero_f32<<<blk(NR128), 256, 0, stream>>>(ACC, NR128);
    edge_scatter128<<<(E_RR + 7) / 8, 256, 0, stream>>>(HBUF, rr_src, rr_dst, DINV, ACC, E_RR, 1);
    gcn_finalize128<<<blk(NR128), 256, 0, stream>>>(ACC, HBUF, DINV, bg2, ACC, N_REGION); // r2 in ACC

    // -------- Layer 2: SAGE region -> subject (x_src = r1, x_dst = s1) --------
    zero_f32<<<blk(NS128), 256, 0, stream>>>(SUM, NS128);
    edge_scatter128<<<(E_HS + 7) / 8, 256, 0, stream>>>(R1, hs_src, hs_dst, nullptr, SUM, E_HS, 0);
    mean_div128<<<blk(NS128), 256, 0, stream>>>(SUM, CNT, N_SUBJECT);
    launch_gemm(SUM, Wl2, bl2, S2, N_SUBJECT, HIDDEN, HIDDEN, 0, stream);   // S2 = HBUF (free now)
    launch_gemm(S1, Wr2, nullptr, S2, N_SUBJECT, HIDDEN, HIDDEN, 1, stream);

    // -------- Output head: shared linear, concat(region, subject) --------
    launch_gemm(ACC, Wout, bout, out, N_REGION, HIDDEN, OUTD, 0, stream);
    launch_gemm(S2, Wout, bout, out + (size_t)N_REGION * OUTD, N_SUBJECT, HIDDEN, OUTD, 0, stream);
}